// EdgeEncoding_82480551953020
// MI455X (gfx1250) — compile-verified
//
#include <hip/hip_runtime.h>

#define TPB   256
#define OUT_F 19     // len(1) + SH l=1,2(8) + cutoff(1) + basis*cutoff(8); o[18] never written
#define RCUT_F 6.0f
#define EPS_F  1e-6f

// Async LDS->global writeback available? (gfx1250 device pass only)
#if defined(__gfx1250__) && __has_builtin(__builtin_amdgcn_global_store_async_from_lds_b128)
#define USE_ASYNC_WB 1
#else
#define USE_ASYNC_WB 0
#endif

// ---------------------------------------------------------------------------
// Kernel 1: combined shift table  T[g][p][j] = sum_i PERIODIC[p][i] * cell[g][i][j]
// 16 graphs x 27 images x 3 comps = 5184 B into d_ws.
// ---------------------------------------------------------------------------
__global__ __launch_bounds__(TPB) void shift_table_kernel(
    const float* __restrict__ cells,   // [ng,3,3]
    float* __restrict__ table,         // [ng*27*3]
    int nent)                          // ng*27
{
    int idx = blockIdx.x * blockDim.x + threadIdx.x;
    if (idx >= nent) return;
    int g = idx / 27;
    int p = idx - g * 27;
    float fi = (float)(p / 9)       - 1.0f;
    float fj = (float)((p / 3) % 3) - 1.0f;
    float fk = (float)(p % 3)       - 1.0f;
    const float* c = cells + g * 9;
#pragma unroll
    for (int j = 0; j < 3; ++j)
        table[idx * 3 + j] = fi * c[j] + fj * c[3 + j] + fk * c[6 + j];
}

#if USE_ASYNC_WB
typedef int v4i __attribute__((vector_size(16)));
typedef v4i __attribute__((address_space(1))) gv4i;   // global
typedef v4i __attribute__((address_space(3))) lv4i;   // LDS

__device__ __forceinline__ void async_store_chunk16(float* gbase, float* ldsbase, int c)
{
    v4i* gp = (v4i*)((char*)gbase   + (size_t)c * 16);
    v4i* lp = (v4i*)((char*)ldsbase + (size_t)c * 16);
    __builtin_amdgcn_global_store_async_from_lds_b128((gv4i*)gp, (lv4i*)lp, 0, 0);
}
#endif

// ---------------------------------------------------------------------------
// Kernel 2: per-edge featurization. One thread per edge; LDS-staged coalesced
// writeback (async LDS->global B128 when available, NT b32 stores otherwise).
// ---------------------------------------------------------------------------
template<bool USE_TABLE>
__global__ __launch_bounds__(TPB) void edge_feat_kernel(
    const float* __restrict__ pos,     // [N,3]
    const float* __restrict__ cells,   // [ng,3,3]
    const int*   __restrict__ ei,      // [2,E] flat
    const int*   __restrict__ pidx,    // [E]
    const int*   __restrict__ batch,   // [N]
    const float* __restrict__ table,   // [ng*27*3] (when USE_TABLE)
    float*       __restrict__ out,     // [E,19]
    int E)
{
    __shared__ __align__(16) float lds[TPB * OUT_F];

    const int tid = threadIdx.x;
    int e  = blockIdx.x * TPB + tid;
    int ec = (e < E) ? e : (E - 1);            // clamp tail lanes (writes bounded)

    // Streaming, never-reused -> non-temporal loads; keep L2 for gather tables.
    int src = __builtin_nontemporal_load(ei + ec);
    int dst = __builtin_nontemporal_load(ei + E + ec);
    int p   = __builtin_nontemporal_load(pidx + ec);

    int g = batch[src];                        // cached (0.4 MB)

    float t0, t1, t2;
    if (USE_TABLE) {
        const float* t = table + (g * 27 + p) * 3;   // cached (5 KB)
        t0 = t[0]; t1 = t[1]; t2 = t[2];
    } else {
        float fi = (float)(p / 9)       - 1.0f;
        float fj = (float)((p / 3) % 3) - 1.0f;
        float fk = (float)(p % 3)       - 1.0f;
        const float* c = cells + g * 9;
        t0 = fi * c[0] + fj * c[3] + fk * c[6];
        t1 = fi * c[1] + fj * c[4] + fk * c[7];
        t2 = fi * c[2] + fj * c[5] + fk * c[8];
    }

    // v = -(pos[dst] - pos[src] - shift@cell) = pos[src] - pos[dst] + T
    float vx = pos[src * 3 + 0] - pos[dst * 3 + 0] + t0;
    float vy = pos[src * 3 + 1] - pos[dst * 3 + 1] + t1;
    float vz = pos[src * 3 + 2] - pos[dst * 3 + 2] + t2;

    float l2  = vx * vx + vy * vy + vz * vz;
    float len = __builtin_amdgcn_sqrtf(l2);          // v_sqrt_f32
    float lm  = fmaxf(len, EPS_F);
    float inv = __builtin_amdgcn_rcpf(lm);           // v_rcp_f32
    float x = vx * inv, y = vy * inv, z = vz * inv;

    float o[OUT_F];
    // --- spherical harmonics lmax=2, component normalization ---
    const float s3  = 1.7320508075688772f;   // sqrt(3)
    const float s5  = 2.2360679774997896f;   // sqrt(5)
    const float s15 = 3.8729833462074170f;   // sqrt(15)
    o[0] = len;
    o[1] = s3 * x;  o[2] = s3 * y;  o[3] = s3 * z;
    o[4] = s15 * x * z;
    o[5] = s15 * x * y;
    o[6] = s5  * (y * y - 0.5f * (x * x + z * z));
    o[7] = s15 * y * z;
    o[8] = 0.5f * s15 * (z * z - x * x);

    // --- polynomial cutoff (P=6): 1 - 28 x^6 + 48 x^7 - 21 x^8 ---
    float xc = fminf(len * (1.0f / RCUT_F), 1.0f);
    float x2 = xc * xc, x4 = x2 * x2, x6 = x4 * x2;
    float cut = 1.0f + x6 * (-28.0f + xc * (48.0f - 21.0f * xc));
    o[9] = cut;

    // --- Bessel: sin(n*pi*safe/6)/(2*safe) * cutoff, n=1..8 ---
    // One v_sin + one v_cos, then Chebyshev recurrence (8 FMAs).
    float safe  = fminf(lm, RCUT_F);
    float theta = safe * 0.5235987755982988f;        // pi/6
    float s1 = __sinf(theta);                        // v_sin_f32
    float c1 = __cosf(theta);                        // v_cos_f32
    float twoc = 2.0f * c1;
    float inv2cut = 0.5f * __builtin_amdgcn_rcpf(safe) * cut;
    float sprev = 0.0f, scur = s1;
#pragma unroll
    for (int n = 0; n < 8; ++n) {
        o[10 + n] = scur * inv2cut;
        float snext = twoc * scur - sprev;
        sprev = scur; scur = snext;
    }

    // --- stage to LDS (stride 19 coprime with 64 banks -> conflict free) ---
#pragma unroll
    for (int j = 0; j < OUT_F; ++j)
        lds[tid * OUT_F + j] = o[j];
    __syncthreads();

    size_t base  = (size_t)blockIdx.x * (TPB * OUT_F);
    size_t total = (size_t)E * OUT_F;

#if USE_ASYNC_WB
    if (base + (size_t)(TPB * OUT_F) <= total) {
        // Full block: stream 19456 B straight from LDS to HBM via async B128
        // stores (ASYNCcnt). No VGPR round-trip. 1216 chunks over 256 lanes.
        constexpr int CHUNKS = (TPB * OUT_F * 4) / 16;        // 1216
        float* gbase = out + base;
#pragma unroll
        for (int r = 0; r < CHUNKS / TPB; ++r)                // 4 full rounds
            async_store_chunk16(gbase, lds, tid + r * TPB);
        if (tid < CHUNKS - (CHUNKS / TPB) * TPB)              // 192-lane tail
            async_store_chunk16(gbase, lds, tid + (CHUNKS / TPB) * TPB);
        // S_ENDPGM performs an implicit wait-idle (covers ASYNCcnt).
        return;
    }
#endif
    // Bounded fallback / tail block: coalesced NT b32 stores.
#pragma unroll
    for (int i = 0; i < OUT_F; ++i) {
        size_t idx = base + (size_t)i * TPB + tid;
        if (idx < total)
            __builtin_nontemporal_store(lds[i * TPB + tid], out + idx);
    }
}

extern "C" void kernel_launch(void* const* d_in, const int* in_sizes, int n_in,
                              void* d_out, int out_size, void* d_ws, size_t ws_size,
                              hipStream_t stream) {
    const float* pos   = (const float*)d_in[0];   // [N,3]
    const float* cells = (const float*)d_in[1];   // [ng,3,3]
    const int*   ei    = (const int*)  d_in[2];   // [2,E]
    const int*   pidx  = (const int*)  d_in[3];   // [E]
    const int*   batch = (const int*)  d_in[4];   // [N]

    int E  = in_sizes[3];
    int ng = in_sizes[1] / 9;
    int nent = ng * 27;
    size_t table_bytes = (size_t)nent * 3 * sizeof(float);

    int nb = (E + TPB - 1) / TPB;

    if (ws_size >= table_bytes) {
        float* table = (float*)d_ws;
        shift_table_kernel<<<(nent + TPB - 1) / TPB, TPB, 0, stream>>>(cells, table, nent);
        edge_feat_kernel<true><<<nb, TPB, 0, stream>>>(pos, cells, ei, pidx, batch, table,
                                                       (float*)d_out, E);
    } else {
        edge_feat_kernel<false><<<nb, TPB, 0, stream>>>(pos, cells, ei, pidx, batch, nullptr,
                                                        (float*)d_out, E);
    }
}